// QLayer_65481071406580
// MI455X (gfx1250) — compile-verified
//
#include <hip/hip_runtime.h>
#include <math.h>

// ---------------------------------------------------------------------------
// Quantum layer, analytically collapsed.
//
//   x_proj = x @ proj_w^T + proj_b                      (256 x 14, f32 GEMM)
//   out[b,i] = prod_{j<=i} cos(sum_k params[j,k]) * cos(x_proj[b,j])
//
// Product state before the CNOT chain; the chain conjugates Z_i into
// Z_0..Z_i; the expectation factorizes into per-qubit cos(P_j)*cos(xp_j).
//
// GEMM path: V_WMMA_F32_16X16X4_F32, one 16-row M tile per wave32,
// N = 16 (cols 14,15 duplicate col 13 — their output is never read),
// K = 128 in 32 chained WMMA steps.  Pure load+WMMA inner loop.
// ---------------------------------------------------------------------------

typedef float v2f __attribute__((ext_vector_type(2)));
typedef float v8f __attribute__((ext_vector_type(8)));

#define N_QUBITS  14
#define N_PARAMS  4
#define INPUT_DIM 128
#define BATCH     256

__global__ __launch_bounds__(32) void qlayer_wmma_kernel(
    const float* __restrict__ x,        // (256, 128)
    const float* __restrict__ proj_w,   // (14, 128)
    const float* __restrict__ proj_b,   // (14,)
    const float* __restrict__ params,   // (14, 4)
    float* __restrict__ out)            // (256, 14)
{
    // one wave32 per block; one 16x16 x_proj tile per wave
    __shared__ float zt[16 * 17];       // stride 17 -> conflict-free row reads

    const int lane = threadIdx.x;       // 0..31
    const int half = lane >> 4;         // K-half of the WMMA fragment
    const int ln   = lane & 15;         // 0..15
    const int tile = blockIdx.x;        // 0..15 (M tile)
    const int row  = tile * 16 + ln;    // A-matrix row for this lane

    // Pad columns 14,15 duplicate qubit 13: WMMA columns are independent,
    // and the epilogue never reads columns >= N_QUBITS. No masking needed.
    const int nclamp = (ln < N_QUBITS) ? ln : (N_QUBITS - 1);

    // A layout (16x4 f32): lane<16 -> K={k0,k0+1}, lane>=16 -> K={k0+2,k0+3}
    // B layout (4x16 f32): same K-half split, N = lane&15; B[k][n] = proj_w[n][k]
    const float* arow = x      + row    * INPUT_DIM + 2 * half;
    const float* brow = proj_w + nclamp * INPUT_DIM + 2 * half;

    v8f acc = {};
#pragma unroll
    for (int kk = 0; kk < INPUT_DIM / 4; ++kk) {
        const int k0 = kk * 4;
        v2f a = *(const v2f*)(arow + k0);
        v2f b = *(const v2f*)(brow + k0);
        // D = A x B + C  (f32 16x16x4): neg_a, A, neg_b, B, c_mod, C, reuse_a, reuse_b
        acc = __builtin_amdgcn_wmma_f32_16x16x4_f32(
            false, a, false, b, (short)0, acc, false, false);
    }

    // Per-lane (per-column n) epilogue constants.
    const float bias = proj_b[nclamp];
    const float P = params[nclamp * N_PARAMS + 0] + params[nclamp * N_PARAMS + 1]
                  + params[nclamp * N_PARAMS + 2] + params[nclamp * N_PARAMS + 3];
    const float cp = __cosf(P);

    // C/D layout: VGPR r holds element (M = half*8 + r, N = ln).
#pragma unroll
    for (int r = 0; r < 8; ++r) {
        const int m = half * 8 + r;
        zt[m * 17 + ln] = __cosf(acc[r] + bias) * cp;   // z[m][n]
    }

    __syncthreads();   // single-wave WG: orders the LDS RAW (s_wait_dscnt)

    // Prefix product along the qubit axis: out[b,i] = prod_{j<=i} z[b][j].
    if (half == 0) {
        const float* zr = zt + ln * 17;
        float* o = out + (tile * 16 + ln) * N_QUBITS;
        float prod = 1.0f;
#pragma unroll
        for (int j = 0; j < N_QUBITS; ++j) {
            prod *= zr[j];
            o[j] = prod;
        }
    }
}

extern "C" void kernel_launch(void* const* d_in, const int* in_sizes, int n_in,
                              void* d_out, int out_size, void* d_ws, size_t ws_size,
                              hipStream_t stream) {
    const float* x      = (const float*)d_in[0];   // (256,128)
    const float* proj_w = (const float*)d_in[1];   // (14,128)
    const float* proj_b = (const float*)d_in[2];   // (14,)
    const float* params = (const float*)d_in[3];   // (14,4)
    float* out = (float*)d_out;                    // (256,14)

    // 16 M-tiles of 16 rows, one wave32 workgroup each -> spread across WGPs
    qlayer_wmma_kernel<<<BATCH / 16, 32, 0, stream>>>(x, proj_w, proj_b, params, out);
}